// USTLayer_5325759447676
// MI455X (gfx1250) — compile-verified
//
#include <hip/hip_runtime.h>

// out[b, l, d] = in[b, l, d] * (l+1)/L   for  [B=8, L=4096, D=2048] f32.
//
// Bandwidth-bound streaming kernel: 512 MiB compulsory HBM traffic,
// 0.125 FLOP/byte -> ~23 us floor at 23.3 TB/s. WMMA/TDM cannot reduce
// compulsory bytes for an elementwise row-scale, so the optimal kernel is
// a pure stream: b128 non-temporal loads/stores, maximal per-wave MLP,
// zero per-access branching.
//
// All control flow is wave-uniform SCALAR (blockIdx-based): the full-tile
// fast path is straight-line code (4 clustered NT b128 loads -> 4 NT b128
// stores); the guarded tail path exists only for robustness and is not
// taken at this problem size. 32-bit index math (n4 = 16M < 2^31) enables
// saddr+voffset addressing instead of 64-bit VALU adds.

typedef float v4f __attribute__((ext_vector_type(4)));

static constexpr unsigned kBlock     = 256;              // 8 wave32 per block
static constexpr unsigned kPerThread = 4;                // float4s per thread
static constexpr unsigned kSpan      = kBlock * kPerThread;  // 1024 float4/block
static constexpr int      kL         = 4096;
static constexpr int      kD4Sh      = 9;                // log2(D/4) = log2(512)
static constexpr float    kInvL      = 1.0f / (float)kL; // exact power of two
static constexpr unsigned kPfBlocks  = 512;              // prefetch 512*16KB = 8 MiB ahead

__global__ __launch_bounds__(kBlock) void scale_rows_kernel(
    const float* __restrict__ in, float* __restrict__ out, unsigned n4) {
  const unsigned base = blockIdx.x * kSpan + threadIdx.x;

  const v4f* __restrict__ in4  = (const v4f*)in;
  v4f* __restrict__       out4 = (v4f*)out;

  // Cross-block prefetch ~8 MiB ahead into GL2 (global_prefetch_b8).
  // Scalar guard: s_cmp + s_cbranch, no exec-mask manipulation.
  if (blockIdx.x + kPfBlocks < gridDim.x) {
    __builtin_prefetch(in + (size_t)(base + kPfBlocks * kSpan) * 4u,
                       /*rw=*/0, /*locality=*/0);
  }

  // Scalar (wave-uniform) full-tile test: all blocks but a possible tail
  // block take the unguarded straight-line path.
  if ((blockIdx.x + 1u) * kSpan <= n4) {
    // Phase 1: 4 NT b128 loads issued back-to-back (clauseable).
    v4f v[kPerThread];
#pragma unroll
    for (unsigned k = 0; k < kPerThread; ++k) {
      v[k] = __builtin_nontemporal_load(in4 + (base + k * kBlock));
    }
    // Phase 2: scale + 4 NT b128 stores.
#pragma unroll
    for (unsigned k = 0; k < kPerThread; ++k) {
      const unsigned i = base + k * kBlock;
      const int   l = (int)(i >> kD4Sh) & (kL - 1);   // (i / (D/4)) % L
      const float s = (float)(l + 1) * kInvL;         // exact in f32
      __builtin_nontemporal_store(v[k] * s, out4 + i);
    }
  } else {
    // Tail block (never taken at B=8, L=4096, D=2048; kept for robustness).
#pragma unroll
    for (unsigned k = 0; k < kPerThread; ++k) {
      const unsigned i = base + k * kBlock;
      if (i < n4) {
        v4f v = __builtin_nontemporal_load(in4 + i);
        const int   l = (int)(i >> kD4Sh) & (kL - 1);
        const float s = (float)(l + 1) * kInvL;
        __builtin_nontemporal_store(v * s, out4 + i);
      }
    }
  }
}

extern "C" void kernel_launch(void* const* d_in, const int* in_sizes, int n_in,
                              void* d_out, int out_size, void* d_ws, size_t ws_size,
                              hipStream_t stream) {
  const float* in = (const float*)d_in[0];
  float* out = (float*)d_out;

  const unsigned n  = (unsigned)in_sizes[0];   // 8*4096*2048 = 67,108,864
  const unsigned n4 = n >> 2;                  // 16,777,216 float4s

  const unsigned grid = (n4 + kSpan - 1) / kSpan;   // 16384 blocks

  scale_rows_kernel<<<grid, kBlock, 0, stream>>>(in, out, n4);
}